// GrossModel_20151986553247
// MI455X (gfx1250) — compile-verified
//
#include <hip/hip_runtime.h>
#include <hip/hip_bf16.h>
#include <math.h>

#define S_  12
#define N_  5000
#define F_  64
#define R_  5
#define E_  256000
#define O1_ 15
#define O2_ 10
#define PAD 16          // padded feature stride for node-feature buffers (64B rows)
#define BN_EPS 1e-5f

typedef __attribute__((ext_vector_type(2))) float v2f;
typedef __attribute__((ext_vector_type(8))) float v8f;

// ---------------------------------------------------------------------------
// utility: fill
// ---------------------------------------------------------------------------
__global__ void fill_f32(float* p, size_t n, float v) {
    size_t i = (size_t)blockIdx.x * blockDim.x + threadIdx.x;
    if (i < n) p[i] = v;
}

// ---------------------------------------------------------------------------
// float atomic max via signed/unsigned int ordering trick
// ---------------------------------------------------------------------------
__device__ __forceinline__ void atomicMaxFloat(float* addr, float val) {
    if (val >= 0.0f) {
        atomicMax((int*)addr, __float_as_int(val));
    } else {
        atomicMin((unsigned int*)addr, __float_as_uint(val));
    }
}

// ---------------------------------------------------------------------------
// vectorized 64B row gather: row must be 16B aligned (it is: 64B records)
// ---------------------------------------------------------------------------
__device__ __forceinline__ void load_row16(const float* __restrict__ row, float* v) {
    const float4* r4 = (const float4*)row;
    float4 a = r4[0], b = r4[1], c = r4[2], d = r4[3];
    v[0]=a.x; v[1]=a.y; v[2]=a.z;  v[3]=a.w;
    v[4]=b.x; v[5]=b.y; v[6]=b.z;  v[7]=b.w;
    v[8]=c.x; v[9]=c.y; v[10]=c.z; v[11]=c.w;
    v[12]=d.x; v[13]=d.y; v[14]=d.z; v[15]=d.w;
}

// ---------------------------------------------------------------------------
// WMMA node transform:  XR[s][r][n][0:16] = X_s[n][0:in_dim] @ W[r] (zero-padded)
// One wave per 16x16 output tile, K swept 4 at a time with
// V_WMMA_F32_16X16X4_F32 (fp32-exact).  A/B staged in LDS with padding.
//
// 32-bit A 16x4 layout : lanes 0-15 -> M=lane, VGPR0=K0 VGPR1=K1
//                        lanes 16-31 -> M=lane-16, VGPR0=K2 VGPR1=K3
// 32-bit B 4x16 layout : mirrored with N in place of M.
// 32-bit C/D 16x16     : VGPR i -> M=i (lanes 0-15) / M=8+i (lanes 16-31), N=lane&15
// ---------------------------------------------------------------------------
__global__ void __launch_bounds__(32)
wmma_node_transform(const float* __restrict__ X, int x_row_stride, size_t x_graph_stride,
                    const float* __restrict__ W,   // [R][in_dim][out_dim]
                    float* __restrict__ XR,        // [S][R][N][PAD]
                    int in_dim, int out_dim) {
    __shared__ float At[16][F_ + 1];   // +1 pad: column reads hit distinct banks
    __shared__ float Bt[F_][17];

    const int s     = blockIdx.z;
    const int r     = blockIdx.y;
    const int tile0 = blockIdx.x * 16;
    const int tid   = threadIdx.x;
    const int kpad  = (in_dim + 3) & ~3;

    const float* Xs = X + (size_t)s * x_graph_stride;
    for (int idx = tid; idx < 16 * kpad; idx += 32) {
        int rr = idx / kpad, cc = idx % kpad;
        int row = tile0 + rr;
        At[rr][cc] = (row < N_ && cc < in_dim) ? Xs[(size_t)row * x_row_stride + cc] : 0.0f;
    }
    const float* Wr = W + (size_t)r * in_dim * out_dim;
    for (int idx = tid; idx < kpad * 16; idx += 32) {
        int kk = idx / 16, nn = idx % 16;
        Bt[kk][nn] = (kk < in_dim && nn < out_dim) ? Wr[(size_t)kk * out_dim + nn] : 0.0f;
    }
    __syncthreads();

    const int m  = tid & 15;          // M for A-frag, N for B-frag / C columns
    const int kb = (tid >> 4) << 1;   // K sub-offset: 0 for lanes 0-15, 2 for 16-31

    v8f c = {};
    for (int k0 = 0; k0 < kpad; k0 += 4) {
        v2f a, b;
        a.x = At[m][k0 + kb];
        a.y = At[m][k0 + kb + 1];
        b.x = Bt[k0 + kb][m];
        b.y = Bt[k0 + kb + 1][m];
        c = __builtin_amdgcn_wmma_f32_16x16x4_f32(false, a, false, b,
                                                  (short)0, c, false, false);
    }

    // pad columns (m >= out_dim) are exact zeros since B was zero-padded:
    // store all 16 columns so downstream b128 gathers see clean data.
    float* out = XR + (((size_t)s * R_ + r) * N_) * PAD;
    const int rbase = (tid >> 4) * 8;
    for (int i = 0; i < 8; ++i) {
        int row = tile0 + rbase + i;
        if (row < N_)
            out[(size_t)row * PAD + m] = c[i];
    }
}

// ---------------------------------------------------------------------------
// Edge pass 1: attention logits + leaky-relu + segment max over dst
// alpha_e = <xr[et,dst],q> + <xr[et,src],k> + ea_e * <lew,e>
// ---------------------------------------------------------------------------
__global__ void edge_alpha(const int* __restrict__ edge_index,  // [S][2][E]
                           const int* __restrict__ edge_type,   // [S][E]
                           const float* __restrict__ edge_attr, // [S][E][1]
                           const float* __restrict__ XR,        // [S][R][N][PAD]
                           const float* __restrict__ q, const float* __restrict__ k,
                           const float* __restrict__ evec, const float* __restrict__ lew,
                           float* __restrict__ ALPHA,           // [S][E]
                           float* __restrict__ AMAX,            // [S][N]
                           int dim) {
    int s = blockIdx.y;
    int e = blockIdx.x * blockDim.x + threadIdx.x;
    if (e >= E_) return;
    const int* ei = edge_index + (size_t)s * 2 * E_;
    int src = ei[e];
    int dst = ei[E_ + e];
    int et  = edge_type[(size_t)s * E_ + e];

    // zero-padded q/k so the full 16-wide dot is exact
    float qv[PAD], kv[PAD];
    for (int d = 0; d < PAD; ++d) {
        qv[d] = (d < dim) ? q[d] : 0.0f;
        kv[d] = (d < dim) ? k[d] : 0.0f;
    }
    float cle = 0.0f;
    for (int d = 0; d < dim; ++d) cle += lew[d] * evec[d];

    float xi[PAD], xj[PAD];
    load_row16(XR + (((size_t)s * R_ + et) * N_ + dst) * PAD, xi);
    load_row16(XR + (((size_t)s * R_ + et) * N_ + src) * PAD, xj);

    float acc = 0.0f;
    #pragma unroll
    for (int d = 0; d < PAD; ++d) acc += xi[d] * qv[d] + xj[d] * kv[d];

    acc += edge_attr[(size_t)s * E_ + e] * cle;
    acc = (acc > 0.0f) ? acc : 0.2f * acc;          // leaky_relu(0.2)
    ALPHA[(size_t)s * E_ + e] = acc;
    atomicMaxFloat(&AMAX[(size_t)s * N_ + dst], acc);
}

// ---------------------------------------------------------------------------
// Edge pass 2: exp(alpha - max) in place + segment sum of denominators
// ---------------------------------------------------------------------------
__global__ void edge_exp(const int* __restrict__ edge_index,
                         const float* __restrict__ AMAX,
                         float* __restrict__ ALPHA,
                         float* __restrict__ DENOM) {
    int s = blockIdx.y;
    int e = blockIdx.x * blockDim.x + threadIdx.x;
    if (e >= E_) return;
    const int* ei = edge_index + (size_t)s * 2 * E_;
    int dst = ei[E_ + e];
    size_t idx = (size_t)s * E_ + e;
    float v = expf(ALPHA[idx] - AMAX[(size_t)s * N_ + dst]);
    ALPHA[idx] = v;
    atomicAdd(&DENOM[(size_t)s * N_ + dst], v);
}

// ---------------------------------------------------------------------------
// Edge pass 3: normalized weight * source features, scatter-add into AGG
// ---------------------------------------------------------------------------
__global__ void edge_aggregate(const int* __restrict__ edge_index,
                               const int* __restrict__ edge_type,
                               const float* __restrict__ XR,
                               const float* __restrict__ AEXP,
                               const float* __restrict__ DENOM,
                               float* __restrict__ AGG,          // [S][N][PAD]
                               int dim) {
    int s = blockIdx.y;
    int e = blockIdx.x * blockDim.x + threadIdx.x;
    if (e >= E_) return;
    const int* ei = edge_index + (size_t)s * 2 * E_;
    int src = ei[e];
    int dst = ei[E_ + e];
    int et  = edge_type[(size_t)s * E_ + e];
    float a = AEXP[(size_t)s * E_ + e] / (DENOM[(size_t)s * N_ + dst] + 1e-16f);

    float xj[PAD];
    load_row16(XR + (((size_t)s * R_ + et) * N_ + src) * PAD, xj);

    float* out = AGG + ((size_t)s * N_ + dst) * PAD;
    for (int d = 0; d < dim; ++d)
        atomicAdd(&out[d], a * xj[d]);
}

// ---------------------------------------------------------------------------
// Node post: +bias, BN(eval), ELU -> H
// ---------------------------------------------------------------------------
__global__ void node_post(const float* __restrict__ AGG, const float* __restrict__ bias,
                          const float* __restrict__ bw, const float* __restrict__ bb,
                          const float* __restrict__ bm, const float* __restrict__ bv,
                          float* __restrict__ H, int dim) {
    int s = blockIdx.y;
    int n = blockIdx.x * blockDim.x + threadIdx.x;
    if (n >= N_) return;
    const float* a = AGG + ((size_t)s * N_ + n) * PAD;
    float* h = H + ((size_t)s * N_ + n) * PAD;
    for (int d = 0; d < dim; ++d) {
        float x = a[d] + bias[d];
        x = (x - bm[d]) * rsqrtf(bv[d] + BN_EPS) * bw[d] + bb[d];
        h[d] = (x > 0.0f) ? x : expm1f(x);          // elu
    }
    for (int d = dim; d < PAD; ++d) h[d] = 0.0f;
}

// ---------------------------------------------------------------------------
// Linear head: y_all[s][n] = <h2[n], lin_w> + lin_b
// ---------------------------------------------------------------------------
__global__ void head_kernel(const float* __restrict__ H2,
                            const float* __restrict__ lin_w, const float* __restrict__ lin_b,
                            float* __restrict__ YALL) {
    int s = blockIdx.y;
    int n = blockIdx.x * blockDim.x + threadIdx.x;
    if (n >= N_) return;
    const float* h = H2 + ((size_t)s * N_ + n) * PAD;
    float acc = lin_b[0];
    for (int d = 0; d < O2_; ++d) acc += h[d] * lin_w[d];
    YALL[(size_t)s * N_ + n] = acc;
}

// ---------------------------------------------------------------------------
// Final: variance proxy, softmax over nodes, clip 0.25, renormalize.
// Single 1024-thread workgroup; N=5000 so each thread handles <=5 nodes.
// ---------------------------------------------------------------------------
__global__ void __launch_bounds__(1024)
final_portfolio(const float* __restrict__ YALL, const float* __restrict__ future_ret,
                const float* __restrict__ gamma, float* __restrict__ Z,
                float* __restrict__ out) {
    __shared__ float red[1024];
    const int tid = threadIdx.x;
    const float g = gamma[0];

    float lmax = -INFINITY;
    for (int n = tid; n < N_; n += 1024) {
        float var = 0.0f;
        for (int s = 0; s < S_ - 1; ++s) {
            float r = future_ret[(size_t)s * N_ + n] - YALL[(size_t)s * N_ + n];
            var += r * r;
        }
        var *= (1.0f / (float)(S_ - 1));
        float z = YALL[(size_t)(S_ - 1) * N_ + n] - g * var;
        Z[n] = z;
        lmax = fmaxf(lmax, z);
    }
    red[tid] = lmax; __syncthreads();
    for (int off = 512; off > 0; off >>= 1) {
        if (tid < off) red[tid] = fmaxf(red[tid], red[tid + off]);
        __syncthreads();
    }
    float zmax = red[0]; __syncthreads();

    float lsum = 0.0f;
    for (int n = tid; n < N_; n += 1024) {
        float e = expf(Z[n] - zmax);
        Z[n] = e;
        lsum += e;
    }
    red[tid] = lsum; __syncthreads();
    for (int off = 512; off > 0; off >>= 1) {
        if (tid < off) red[tid] += red[tid + off];
        __syncthreads();
    }
    float tot = red[0]; __syncthreads();

    float lsum2 = 0.0f;
    for (int n = tid; n < N_; n += 1024) {
        float w = Z[n] / tot;
        w = fminf(fmaxf(w, 0.0f), 0.25f);
        Z[n] = w;
        lsum2 += w;
    }
    red[tid] = lsum2; __syncthreads();
    for (int off = 512; off > 0; off >>= 1) {
        if (tid < off) red[tid] += red[tid + off];
        __syncthreads();
    }
    float tot2 = red[0]; __syncthreads();

    for (int n = tid; n < N_; n += 1024)
        out[n] = Z[n] / tot2;
}

// ---------------------------------------------------------------------------
// host-side driver
// ---------------------------------------------------------------------------
extern "C" void kernel_launch(void* const* d_in, const int* in_sizes, int n_in,
                              void* d_out, int out_size, void* d_ws, size_t ws_size,
                              hipStream_t stream) {
    (void)in_sizes; (void)n_in; (void)out_size; (void)ws_size;

    const float* x          = (const float*)d_in[0];
    const float* edge_attr  = (const float*)d_in[1];
    const float* future_ret = (const float*)d_in[2];
    const float* w1   = (const float*)d_in[3];
    const float* q1   = (const float*)d_in[4];
    const float* k1   = (const float*)d_in[5];
    const float* e1   = (const float*)d_in[6];
    const float* lew1 = (const float*)d_in[7];
    const float* b1   = (const float*)d_in[8];
    const float* w2   = (const float*)d_in[9];
    const float* q2   = (const float*)d_in[10];
    const float* k2   = (const float*)d_in[11];
    const float* e2   = (const float*)d_in[12];
    const float* lew2 = (const float*)d_in[13];
    const float* b2   = (const float*)d_in[14];
    const float* bn1_w = (const float*)d_in[15];
    const float* bn1_b = (const float*)d_in[16];
    const float* bn1_m = (const float*)d_in[17];
    const float* bn1_v = (const float*)d_in[18];
    const float* bn2_w = (const float*)d_in[19];
    const float* bn2_b = (const float*)d_in[20];
    const float* bn2_m = (const float*)d_in[21];
    const float* bn2_v = (const float*)d_in[22];
    const float* lin_w = (const float*)d_in[23];
    const float* lin_b = (const float*)d_in[24];
    const float* gamma = (const float*)d_in[25];
    const int* edge_index = (const int*)d_in[26];
    const int* edge_type  = (const int*)d_in[27];

    // workspace carve-out (floats); every region is a multiple of 16 floats,
    // so all PAD-stride rows stay 64B aligned for b128 gathers.
    float* ws = (float*)d_ws;
    size_t off = 0;
    float* XR    = ws + off; off += (size_t)S_ * R_ * N_ * PAD;   // 4.80M
    float* ALPHA = ws + off; off += (size_t)S_ * E_;              // 3.07M
    float* AMAX  = ws + off; off += (size_t)S_ * N_;
    float* DENOM = ws + off; off += (size_t)S_ * N_;
    float* AGG   = ws + off; off += (size_t)S_ * N_ * PAD;
    float* H1    = ws + off; off += (size_t)S_ * N_ * PAD;
    float* H2    = ws + off; off += (size_t)S_ * N_ * PAD;
    float* YALL  = ws + off; off += (size_t)S_ * N_;
    float* ZBUF  = ws + off; off += (size_t)N_;

    const dim3 eb(256);
    const dim3 eg((E_ + 255) / 256, S_);
    const dim3 ng((N_ + 255) / 256, S_);
    const dim3 tg((N_ + 15) / 16, R_, S_);

    const size_t snCount  = (size_t)S_ * N_;
    const size_t aggCount = (size_t)S_ * N_ * PAD;
    const unsigned fbN  = (unsigned)((snCount + 255) / 256);
    const unsigned fbA  = (unsigned)((aggCount + 255) / 256);

    struct Layer {
        const float *Xin; int x_rs; size_t x_gs;
        const float *W, *q, *k, *e, *lew, *bias;
        const float *bw, *bb, *bm, *bv;
        float *H; int in_dim, out_dim;
    } layers[2] = {
        { x,  F_,  (size_t)N_ * F_,  w1, q1, k1, e1, lew1, b1,
          bn1_w, bn1_b, bn1_m, bn1_v, H1, F_,  O1_ },
        { H1, PAD, (size_t)N_ * PAD, w2, q2, k2, e2, lew2, b2,
          bn2_w, bn2_b, bn2_m, bn2_v, H2, O1_, O2_ },
    };

    for (int l = 0; l < 2; ++l) {
        const Layer& L = layers[l];
        fill_f32<<<fbN, 256, 0, stream>>>(AMAX,  snCount,  -INFINITY);
        fill_f32<<<fbN, 256, 0, stream>>>(DENOM, snCount,  0.0f);
        fill_f32<<<fbA, 256, 0, stream>>>(AGG,   aggCount, 0.0f);

        wmma_node_transform<<<tg, 32, 0, stream>>>(L.Xin, L.x_rs, L.x_gs,
                                                   L.W, XR, L.in_dim, L.out_dim);
        edge_alpha<<<eg, eb, 0, stream>>>(edge_index, edge_type, edge_attr, XR,
                                          L.q, L.k, L.e, L.lew, ALPHA, AMAX, L.out_dim);
        edge_exp<<<eg, eb, 0, stream>>>(edge_index, AMAX, ALPHA, DENOM);
        edge_aggregate<<<eg, eb, 0, stream>>>(edge_index, edge_type, XR, ALPHA, DENOM,
                                              AGG, L.out_dim);
        node_post<<<ng, eb, 0, stream>>>(AGG, L.bias, L.bw, L.bb, L.bm, L.bv,
                                         L.H, L.out_dim);
    }

    head_kernel<<<ng, eb, 0, stream>>>(H2, lin_w, lin_b, YALL);
    final_portfolio<<<1, 1024, 0, stream>>>(YALL, future_ret, gamma, ZBUF, (float*)d_out);
}